// GaussianRenderer_34059090657616
// MI455X (gfx1250) — compile-verified
//
#include <hip/hip_runtime.h>
#include <hip/hip_bf16.h>

#define IMG_H 128
#define IMG_W 128
#define KTOP  10
#define NG    2048

typedef __attribute__((ext_vector_type(2))) float v2f;
typedef __attribute__((ext_vector_type(8))) float v8f;

// ---------------------------------------------------------------------------
// Phase 1: per-gaussian bilinear-form coefficients over basis
//   [px^2, px*py, py^2, px, py, 1, 0, 0]   (padded to 8 for K=4+4 WMMA;
//   the zero pad also serves as the A-tile source for lanes 16..31 of the
//   second WMMA, so no divergent load is needed in the hot loop)
// q(p,g) = basis(p) . coeff(g)
// ---------------------------------------------------------------------------
__global__ void prep_coeffs(const float* __restrict__ means,      // NG x 2
                            const float* __restrict__ rots,       // NG
                            const float* __restrict__ logscales,  // NG x 2
                            float* __restrict__ coeffs)           // NG x 8
{
    int g = blockIdx.x * blockDim.x + threadIdx.x;
    if (g >= NG) return;
    float mx = means[2 * g], my = means[2 * g + 1];
    float t  = rots[g];
    float ct = __cosf(t), st = __sinf(t);
    float ivx = __expf(-2.0f * logscales[2 * g]);
    float ivy = __expf(-2.0f * logscales[2 * g + 1]);
    float a = ct * ct * ivx + st * st * ivy;
    float c = st * st * ivx + ct * ct * ivy;
    float b = ct * st * (ivx - ivy);
    float* o = coeffs + g * 8;
    o[0] = a;
    o[1] = 2.0f * b;
    o[2] = c;
    o[3] = -2.0f * (a * mx + b * my);
    o[4] = -2.0f * (b * mx + c * my);
    o[5] = a * mx * mx + 2.0f * b * mx * my + c * my * my;
    o[6] = 0.0f;
    o[7] = 0.0f;
}

// ---------------------------------------------------------------------------
// Phase 2: one wave32 renders 16 pixels.
//   A (16 gaussians x K=8 coeffs) x B (K=8 x 16 pixels) via two
//   v_wmma_f32_16x16x4_f32 chained on the accumulator.
//   D layout: lane n (n<16) VGPR r = q(gaussian g0+r,   pixel n)
//             lane n+16     VGPR r = q(gaussian g0+8+r, pixel n)
//   Each lane keeps a sorted top-10 of its gaussian slice, merged with
//   lane^16 via shfl_xor at the end, then alpha-composited.
// ---------------------------------------------------------------------------
__global__ void __launch_bounds__(256) splat_kernel(
    const float* __restrict__ coeffs,   // NG x 8
    const float* __restrict__ colors,   // NG x 3
    float* __restrict__ out)            // H x W x 3
{
    const int lane   = threadIdx.x & 31;
    const int wave   = threadIdx.x >> 5;
    const int waveId = blockIdx.x * (blockDim.x >> 5) + wave;
    const int n      = lane & 15;        // pixel slot within wave
    const bool hi    = lane >= 16;
    const int p      = waveId * 16 + n;  // linear pixel index (row-major)
    const float px   = (float)(p & (IMG_W - 1)) + 0.5f;
    const float py   = (float)(p >> 7) + 0.5f;

    // B matrix (8 x 16 basis, split as two K=4 tiles), constant for the loop.
    // 32-bit B layout: lane n holds {K0,K1} of column n, lane n+16 {K2,K3}.
    v2f B1, B2;
    if (!hi) { B1.x = px * px; B1.y = px * py; B2.x = py;   B2.y = 1.0f; }
    else     { B1.x = py * py; B1.y = px;      B2.x = 0.0f; B2.y = 0.0f; }

    float ta[KTOP];  // descending alphas
    int   ti[KTOP];  // matching gaussian indices
#pragma unroll
    for (int k = 0; k < KTOP; ++k) { ta[k] = -1.0f; ti[k] = 0; }

    const v2f* cf = (const v2f*)coeffs;  // 4 x v2f per gaussian
    // Per-lane v2f slot offsets within a gaussian's 4 x v2f record:
    //   lo lanes: A1 = {c0,c1} (slot 0), A2 = {c4,c5} (slot 2)
    //   hi lanes: A1 = {c2,c3} (slot 1), A2 = {0,0}   (slot 3, zero pad)
    const int slotA1 = hi ? 1 : 0;
    const int slotA2 = hi ? 3 : 2;

#pragma unroll 2
    for (int g0 = 0; g0 < NG; g0 += 16) {
        const v2f* gp = cf + (size_t)(g0 + n) * 4;
        v2f A1 = gp[slotA1];
        v2f A2 = gp[slotA2];

        v8f acc = {};
        acc = __builtin_amdgcn_wmma_f32_16x16x4_f32(
                  false, A1, false, B1, (short)0, acc, false, false);
        acc = __builtin_amdgcn_wmma_f32_16x16x4_f32(
                  false, A2, false, B2, (short)0, acc, false, false);

        const int gbase = g0 + (hi ? 8 : 0);
#pragma unroll
        for (int r = 0; r < 8; ++r) {
            float q = acc[r];
            // exp(-0.5*q) == exp2(-0.5*log2(e)*q).  Raw v_exp_f32: inputs
            // below -126 flush to 0, which is the desired alpha underflow.
            float alpha = __builtin_amdgcn_exp2f(-0.72134752044f * q);
            int   gi    = gbase + r;
            if (alpha > ta[KTOP - 1]) {           // beats current minimum
                float ca = alpha; int ci = gi;    // branchless sorted insert
#pragma unroll
                for (int k = 0; k < KTOP; ++k) {
                    bool  gt = ca > ta[k];
                    float sa = ta[k]; int si = ti[k];
                    ta[k] = gt ? ca : sa;  ti[k] = gt ? ci : si;
                    ca    = gt ? sa : ca;  ci    = gt ? si : ci;
                }
            }
        }
    }

    // Merge with partner lane (lane ^ 16): insert its 10 sorted entries.
#pragma unroll
    for (int k = 0; k < KTOP; ++k) {
        float ca = __shfl_xor(ta[k], 16, 32);
        int   ci = __shfl_xor(ti[k], 16, 32);
#pragma unroll
        for (int j = 0; j < KTOP; ++j) {
            bool  gt = ca > ta[j];
            float sa = ta[j]; int si = ti[j];
            ta[j] = gt ? ca : sa;  ti[j] = gt ? ci : si;
            ca    = gt ? sa : ca;  ci    = gt ? si : ci;
        }
    }

    // Front-to-back composite (list already sorted descending, matching
    // top_k order + cumprod transmittance in the reference).
    float T = 1.0f, rr = 0.0f, gg = 0.0f, bb = 0.0f;
#pragma unroll
    for (int k = 0; k < KTOP; ++k) {
        float a = ta[k] > 0.0f ? ta[k] : 0.0f;
        const float* cp = colors + (size_t)ti[k] * 3;
        float w = a * T;
        rr += w * cp[0];
        gg += w * cp[1];
        bb += w * cp[2];
        T *= (1.0f - a);
    }
    if (!hi) {
        float* op = out + (size_t)p * 3;
        op[0] = rr; op[1] = gg; op[2] = bb;
    }
}

// ---------------------------------------------------------------------------
extern "C" void kernel_launch(void* const* d_in, const int* in_sizes, int n_in,
                              void* d_out, int out_size, void* d_ws, size_t ws_size,
                              hipStream_t stream) {
    const float* means     = (const float*)d_in[0];
    const float* rots      = (const float*)d_in[1];
    const float* logscales = (const float*)d_in[2];
    const float* colors    = (const float*)d_in[3];
    float* coeffs = (float*)d_ws;     // NG * 8 floats = 64 KB scratch
    float* out    = (float*)d_out;    // 128*128*3 floats

    prep_coeffs<<<NG / 256, 256, 0, stream>>>(means, rots, logscales, coeffs);

    // 16384 pixels / 16 per wave = 1024 waves; 8 waves (256 thr) per block.
    splat_kernel<<<(IMG_H * IMG_W) / (16 * 8), 256, 0, stream>>>(coeffs, colors, out);
}